// VandermondeKernel_14774687498510
// MI455X (gfx1250) — compile-verified
//
#include <hip/hip_runtime.h>

typedef __attribute__((ext_vector_type(2))) float v2f;
typedef __attribute__((ext_vector_type(8))) float v8f;

#define P_DIM 256
#define H_DIM 512
#define L_DIM 32768
#define K_DIM 512               // 2*P (real GEMM K)
#define NCOLS (2 * L_DIM)       // interleaved re/im columns
#define MT 128
#define NT 128
#define KT 32
#define LPAD 4

// ---------------------------------------------------------------------------
// Prep 1: polar decomposition of lambda (f64 for phase accuracy at l up to 32768)
// ---------------------------------------------------------------------------
__global__ void vand_polar_prep(const float* __restrict__ lam,
                                double* __restrict__ polar) {
  int p = threadIdx.x;
  if (p < P_DIM) {
    double re = (double)lam[2 * p];
    double im = (double)lam[2 * p + 1];
    polar[2 * p]     = 0.5 * log(re * re + im * im);  // log|lambda|
    polar[2 * p + 1] = atan2(im, re);                 // arg(lambda)
  }
}

// ---------------------------------------------------------------------------
// Prep 2: A[h,p] = C[h,p] * B[p,h] (complex), stored K-major (512 x 512 f32).
// K rows interleave Re/Im in 32-row groups: row (p/16)*32 + (p%16)      = Re
//                                           row (p/16)*32 + 16 + (p%16) = Im
// so a 32-row K-chunk covers 16 p's with both parts (one transcendental
// evaluation in the GEMM feeds both the Re and Im rows).
// ---------------------------------------------------------------------------
__global__ void vand_cb_prep(const float* __restrict__ Bb,
                             const float* __restrict__ Cm,
                             float* __restrict__ A) {
  int idx = blockIdx.x * blockDim.x + threadIdx.x;   // h*P + p
  if (idx >= H_DIM * P_DIM) return;
  int h = idx >> 8;
  int p = idx & (P_DIM - 1);
  float cre = Cm[2 * idx], cim = Cm[2 * idx + 1];
  int b = p * H_DIM + h;
  float bre = Bb[2 * b], bim = Bb[2 * b + 1];
  int rowRe = ((p >> 4) << 5) + (p & 15);
  A[(size_t)rowRe * H_DIM + h]        = cre * bre - cim * bim;  // Re(CB)
  A[(size_t)(rowRe + 16) * H_DIM + h] = cre * bim + cim * bre;  // Im(CB)
}

// ---------------------------------------------------------------------------
// Main: D(512 x 65536) = A(512 x 512) x B(512 x 65536) with B (Vandermonde)
// synthesized per K-chunk in LDS. v_wmma_f32_16x16x4_f32 accumulators.
// Grid: (65536/NT, 512/MT) = (512, 4); block = 256 threads = 8 waves.
// Wave tiling: 4 (M) x 2 (N); each wave owns 32 x 64 of D = 2x4 v8f accs.
// ---------------------------------------------------------------------------
__global__ __launch_bounds__(256)
void vand_wmma_gemm(const float* __restrict__ A,
                    const double* __restrict__ polar,
                    float* __restrict__ out) {
  __shared__ float As[KT][MT + LPAD];   // A chunk, [k][m]
  __shared__ float Bs[KT][NT + LPAD];   // B chunk, [k][n]

  const int tid   = threadIdx.x;
  const int lane  = tid & 31;
  const int wave  = tid >> 5;
  const int waveM = wave & 3;           // 32-row strip
  const int waveN = wave >> 2;          // 64-col strip
  const int lo    = lane & 15;
  const int hi    = lane >> 4;          // 0: K={0,1}/M rows 0-15 ; 1: K={2,3}

  const int mBase = blockIdx.y * MT;
  const int nBase = blockIdx.x * NT;    // column in interleaved (2L) space
  const int lBase = nBase >> 1;         // 64 l's per tile

  v8f acc[2][4];
#pragma unroll
  for (int i = 0; i < 2; ++i)
#pragma unroll
    for (int j = 0; j < 4; ++j)
#pragma unroll
      for (int r = 0; r < 8; ++r) acc[i][j][r] = 0.0f;

  const double TWO_PI = 6.283185307179586476925287;

  for (int kc = 0; kc < K_DIM / KT; ++kc) {
    // ---- stage A chunk (K-major in global) into LDS: 4096 f32, float4 each
#pragma unroll
    for (int i = 0; i < 4; ++i) {
      int q  = tid + i * 256;                       // float4 id in [0,1024)
      int kl = q >> 5;                              // 32 float4 per 128-f row
      int cm = (q & 31) << 2;
      const float* src = A + (size_t)(kc * KT + kl) * H_DIM + mBase + cm;
      *(float4*)&As[kl][cm] = *(const float4*)src;
      if (kc + 1 < K_DIM / KT)
        __builtin_prefetch(src + KT * H_DIM, 0, 1); // next chunk -> L2
    }

    // ---- synthesize Vandermonde B chunk: 16 p's x 64 l's, both complex parts
    const int pBase = kc * 16;
#pragma unroll
    for (int i = 0; i < 4; ++i) {
      int q  = tid + i * 256;                       // pair id in [0,1024)
      int pl = q >> 6;                              // 0..15
      int cl = q & 63;                              // 0..63
      int p  = pBase + pl;
      double ld = (double)(lBase + cl);
      double lr = polar[2 * p];
      double th = polar[2 * p + 1];
      float  w  = __expf((float)(ld * lr));         // r^l (underflows to 0 ok)
      double ph = ld * th;
      ph -= floor(ph * (1.0 / TWO_PI)) * TWO_PI;    // exact-ish range reduce
      float s, c;
      __sincosf((float)ph, &s, &c);
      float vre = w * c;
      float vim = w * s;
      Bs[pl][2 * cl]          =  vre;   // Re(CB) rows: (Vre, Vim)
      Bs[pl][2 * cl + 1]      =  vim;
      Bs[pl + 16][2 * cl]     = -vim;   // Im(CB) rows: (-Vim, Vre)
      Bs[pl + 16][2 * cl + 1] =  vre;
    }
    __syncthreads();

    // ---- 8 k-steps of v_wmma_f32_16x16x4_f32 over the 32-deep chunk
#pragma unroll
    for (int ks = 0; ks < KT / 4; ++ks) {
      const int kb = ks * 4 + 2 * hi;
      v2f a[2], b[4];
#pragma unroll
      for (int i = 0; i < 2; ++i) {                 // A frags: 16x4, [m][k]
        int m = waveM * 32 + i * 16 + lo;
        a[i].x = As[kb][m];
        a[i].y = As[kb + 1][m];
      }
#pragma unroll
      for (int j = 0; j < 4; ++j) {                 // B frags: 4x16, [k][n]
        int n = waveN * 64 + j * 16 + lo;
        b[j].x = Bs[kb][n];
        b[j].y = Bs[kb + 1][n];
      }
#pragma unroll
      for (int i = 0; i < 2; ++i)
#pragma unroll
        for (int j = 0; j < 4; ++j)
          acc[i][j] = __builtin_amdgcn_wmma_f32_16x16x4_f32(
              false, a[i], false, b[j], (short)0, acc[i][j], false, false);
    }
    __syncthreads();
  }

  // ---- store D (already in interleaved complex64 layout)
  // C/D layout: n = lo, m = r + 8*hi within each 16x16 tile
#pragma unroll
  for (int i = 0; i < 2; ++i) {
#pragma unroll
    for (int j = 0; j < 4; ++j) {
      const int col     = nBase + waveN * 64 + j * 16 + lo;
      const int rowBase = mBase + waveM * 32 + i * 16 + hi * 8;
#pragma unroll
      for (int r = 0; r < 8; ++r)
        out[(size_t)(rowBase + r) * NCOLS + col] = acc[i][j][r];
    }
  }
}

// ---------------------------------------------------------------------------
extern "C" void kernel_launch(void* const* d_in, const int* in_sizes, int n_in,
                              void* d_out, int out_size, void* d_ws,
                              size_t ws_size, hipStream_t stream) {
  (void)in_sizes; (void)n_in; (void)out_size; (void)ws_size;
  const float* lam = (const float*)d_in[0];   // Lambda_bar (P,2)
  const float* Bb  = (const float*)d_in[1];   // B_bar (P,H,2)
  const float* Cm  = (const float*)d_in[2];   // C (H,P,2)
  float* out = (float*)d_out;                 // (H,L) complex64 -> f32 pairs

  double* polar = (double*)d_ws;                      // 4 KiB
  float*  A     = (float*)((char*)d_ws + 8192);       // 1 MiB packed CB

  vand_polar_prep<<<1, 256, 0, stream>>>(lam, polar);
  vand_cb_prep<<<(H_DIM * P_DIM) / 256, 256, 0, stream>>>(Bb, Cm, A);

  dim3 grid(NCOLS / NT, H_DIM / MT);   // (512, 4)
  vand_wmma_gemm<<<grid, 256, 0, stream>>>(A, polar, out);
}